// GINConv_32487132627458
// MI455X (gfx1250) — compile-verified
//
#include <hip/hip_runtime.h>

#define DIM       128
#define TILE_M    16
#define MLP_WAVES 4
#define ROW_PAD   4                       // pad h rows to 132 dwords -> conflict-free A frags
#define WSTEPS    (DIM / 4)               // 32 k-steps per GEMM
#define WENTRIES  (8 * WSTEPS * 32)       // [nt][step][lane] float2 entries = 64KB

typedef float v2f __attribute__((ext_vector_type(2)));
typedef float v8f __attribute__((ext_vector_type(8)));

// -------------------------------------------------------------------------
// Phase 1: edge scatter-add.  One wave32 per edge; src/dst loads are wave-
// uniform (scalarized -> s_load), each lane moves a float4 (16B) of the
// 512B row and performs 4 global f32 atomics into agg[dst].  x (51MB) and
// agg (51MB) are both L2-resident on MI455X (192MB L2), so the 819MB of
// gathered reads and the 205M atomics run at L2 speed, not HBM.
// -------------------------------------------------------------------------
__global__ void __launch_bounds__(256) gin_scatter_kernel(
    const float* __restrict__ x,
    const int*   __restrict__ src,
    const int*   __restrict__ dst,
    float*       __restrict__ agg,
    int n_edges)
{
    const int wid  = (int)(blockIdx.x * (blockDim.x >> 5)) + (int)(threadIdx.x >> 5);
    const int lane = threadIdx.x & 31;
    if (wid >= n_edges) return;

    const int s = src[wid];   // wave-uniform -> scalar load
    const int d = dst[wid];   // wave-uniform -> scalar load

    const float4 v = reinterpret_cast<const float4*>(x + (size_t)s * DIM)[lane];
    float* p = agg + (size_t)d * DIM + (size_t)lane * 4;
    atomicAdd(p + 0, v.x);
    atomicAdd(p + 1, v.y);
    atomicAdd(p + 2, v.z);
    atomicAdd(p + 3, v.w);
}

// -------------------------------------------------------------------------
// Phase 2: fused MLP  out = relu((x+agg) @ W1 + b1) @ W2 + b2
// One wave32 per 16-row tile, V_WMMA_F32_16X16X4_F32 (fp32-exact vs ref).
//
// Fragment layouts (ISA 7.12.2, wave32):
//   A 16x4 f32 : lane L holds row M=L&15, K = (L>>4)*2 + {0,1}
//   B 4x16 f32 : lane L holds col N=L&15, K = (L>>4)*2 + {0,1}
//   C/D 16x16  : lane L holds col N=L&15, VGPR v = row M = v + (L>>4)*8
//
// Both operands are LDS-fed:
//   - weights are staged per-block into LDS in fragment order, so the B
//     fetch is ONE ds_load_b64 at (base + lane*8): 64 banks, conflict-free
//   - h rows padded to 132 dwords: A ds_load_b64 hits banks 4L+{0..3},
//     conflict-free across the wave
// -------------------------------------------------------------------------
__global__ void __launch_bounds__(MLP_WAVES * 32) gin_mlp_kernel(
    const float* __restrict__ x,
    const float* __restrict__ agg,
    const float* __restrict__ W1,
    const float* __restrict__ b1,
    const float* __restrict__ W2,
    const float* __restrict__ b2,
    float*       __restrict__ out,
    int n_nodes)
{
    __shared__ float2 wl[WENTRIES];                        // 64 KB, reused W1 -> W2
    __shared__ float  sh[MLP_WAVES][TILE_M][DIM + ROW_PAD]; // ~33 KB, wave-private slices

    const int wave = threadIdx.x >> 5;
    const int lane = threadIdx.x & 31;
    const int tile = blockIdx.x * MLP_WAVES + wave;
    const int row0 = tile * TILE_M;
    const bool valid = (row0 < n_nodes);
    const int mmax = valid ? min(TILE_M, n_nodes - row0) : 0;

    // ---- stage h = x + agg into this wave's LDS slice (float4 traffic) ----
    if (valid) {
        for (int i = lane; i < TILE_M * (DIM / 4); i += 32) {
            const int m  = i >> 5;          // DIM/4 == 32 float4 per row
            const int c4 = i & 31;
            float4 h;
            if (m < mmax) {
                const size_t r = (size_t)(row0 + m) * DIM;
                const float4 xv = reinterpret_cast<const float4*>(x + r)[c4];
                const float4 av = reinterpret_cast<const float4*>(agg + r)[c4];
                h = make_float4(xv.x + av.x, xv.y + av.y, xv.z + av.z, xv.w + av.w);
            } else {
                h = make_float4(0.f, 0.f, 0.f, 0.f);
            }
            float* row = &sh[wave][m][0];
            row[c4 * 4 + 0] = h.x; row[c4 * 4 + 1] = h.y;
            row[c4 * 4 + 2] = h.z; row[c4 * 4 + 3] = h.w;
        }
    }

    // ---- stage W1 into LDS in fragment order (whole block cooperates) ----
    for (int idx = threadIdx.x; idx < WENTRIES; idx += MLP_WAVES * 32) {
        const int nt = idx >> 10;             // / (WSTEPS*32)
        const int s  = (idx >> 5) & 31;       // k-step
        const int L  = idx & 31;              // consumer lane
        const int k  = (s << 2) + ((L >> 4) << 1);
        const int n  = (nt << 4) + (L & 15);
        wl[idx] = make_float2(W1[(size_t)k * DIM + n],
                              W1[(size_t)(k + 1) * DIM + n]);
    }
    __syncthreads();

    const int m_a  = lane & 15;          // A-fragment row
    const int koff = (lane >> 4) * 2;    // A-fragment K sub-offset
    const int n_l  = lane & 15;          // C column
    const int mb   = (lane >> 4) * 8;    // C row base
    const float* hrow = &sh[wave][m_a][0];

    // ---------------- GEMM1: h @ W1 (keep all 8 N-tiles in VGPRs) ---------
    v8f acc[8];
    if (valid) {
        #pragma unroll
        for (int nt = 0; nt < 8; ++nt) {
            v8f c = {0.f, 0.f, 0.f, 0.f, 0.f, 0.f, 0.f, 0.f};
            const float2* wf = wl + (nt << 10) + lane;
            for (int s = 0; s < WSTEPS; ++s) {
                v2f a, b;
                a.x = hrow[(s << 2) + koff];       // ds_load_b64 (conflict-free)
                a.y = hrow[(s << 2) + koff + 1];
                const float2 bb = wf[s << 5];      // ds_load_b64 (conflict-free)
                b.x = bb.x; b.y = bb.y;
                c = __builtin_amdgcn_wmma_f32_16x16x4_f32(
                        false, a, false, b, (short)0, c, false, false);
            }
            acc[nt] = c;
        }

        // bias + ReLU, overwrite own LDS slice (same-wave DS ops in-order)
        #pragma unroll
        for (int nt = 0; nt < 8; ++nt) {
            const int col  = nt * 16 + n_l;
            const float bv = b1[col];
            #pragma unroll
            for (int v = 0; v < 8; ++v) {
                const float t = acc[nt][v] + bv;
                sh[wave][mb + v][col] = t > 0.f ? t : 0.f;
            }
        }
    }
    __syncthreads();   // all waves done reading W1 fragments

    // ---- stage W2 into the same LDS buffer --------------------------------
    for (int idx = threadIdx.x; idx < WENTRIES; idx += MLP_WAVES * 32) {
        const int nt = idx >> 10;
        const int s  = (idx >> 5) & 31;
        const int L  = idx & 31;
        const int k  = (s << 2) + ((L >> 4) << 1);
        const int n  = (nt << 4) + (L & 15);
        wl[idx] = make_float2(W2[(size_t)k * DIM + n],
                              W2[(size_t)(k + 1) * DIM + n]);
    }
    __syncthreads();

    // ---------------- GEMM2: h2 @ W2 + b2 -> out ---------------------------
    if (valid) {
        #pragma unroll
        for (int nt = 0; nt < 8; ++nt) {
            v8f c = {0.f, 0.f, 0.f, 0.f, 0.f, 0.f, 0.f, 0.f};
            const float2* wf = wl + (nt << 10) + lane;
            for (int s = 0; s < WSTEPS; ++s) {
                v2f a, b;
                a.x = hrow[(s << 2) + koff];
                a.y = hrow[(s << 2) + koff + 1];
                const float2 bb = wf[s << 5];
                b.x = bb.x; b.y = bb.y;
                c = __builtin_amdgcn_wmma_f32_16x16x4_f32(
                        false, a, false, b, (short)0, c, false, false);
            }
            const int col  = nt * 16 + n_l;
            const float bv = b2[col];
            #pragma unroll
            for (int v = 0; v < 8; ++v) {
                const int m = mb + v;
                if (m < mmax)
                    out[(size_t)(row0 + m) * DIM + col] = c[v] + bv;
            }
        }
    }
}

// -------------------------------------------------------------------------
// Host-side launcher (graph-capture safe: only async ops on `stream`)
// -------------------------------------------------------------------------
extern "C" void kernel_launch(void* const* d_in, const int* in_sizes, int n_in,
                              void* d_out, int out_size, void* d_ws, size_t ws_size,
                              hipStream_t stream) {
    const float* x  = (const float*)d_in[0];
    const int*   ei = (const int*)  d_in[1];   // edge_index, shape (2, E) row-major
    const float* W1 = (const float*)d_in[2];
    const float* b1 = (const float*)d_in[3];
    const float* W2 = (const float*)d_in[4];
    const float* b2 = (const float*)d_in[5];
    float* out = (float*)d_out;

    const int n_nodes = in_sizes[0] / DIM;     // 100000
    const int n_edges = in_sizes[1] / 2;       // 1600000

    float* agg = (float*)d_ws;                 // [n_nodes, DIM] scratch

    // zero the aggregation buffer (memset is graph-capturable)
    hipMemsetAsync(agg, 0, (size_t)n_nodes * DIM * sizeof(float), stream);

    // one wave32 per edge, 8 waves per 256-thread block
    const int blocks_sc = (n_edges + 7) / 8;
    gin_scatter_kernel<<<blocks_sc, 256, 0, stream>>>(
        x, ei, ei + n_edges, agg, n_edges);

    // one wave32 per 16-row tile, 4 waves per block
    const int tiles      = (n_nodes + TILE_M - 1) / TILE_M;
    const int blocks_mlp = (tiles + MLP_WAVES - 1) / MLP_WAVES;
    gin_mlp_kernel<<<blocks_mlp, MLP_WAVES * 32, 0, stream>>>(
        x, agg, W1, b1, W2, b2, out, n_nodes);
}